// MTF_68461778698457
// MI455X (gfx1250) — compile-verified
//
#include <hip/hip_runtime.h>

#define LSER 256
#define NB   65
#define NC   (NB * NB)   // 4225

typedef __attribute__((address_space(3))) float lds_f32;

__global__ __launch_bounds__(256) void mtf_kernel(const float* __restrict__ x,
                                                  float* __restrict__ out) {
    __shared__ float xs[LSER];
    __shared__ alignas(16) unsigned long long keys[LSER];
    __shared__ unsigned char binsS[LSER];
    __shared__ unsigned int counts[NC];
    __shared__ float lut[LSER];          // lut[c] = (float)c / norm
    __shared__ int sse[2];               // [0]=start (min), [1]=end (max)

    const int series = blockIdx.x;
    const int t = threadIdx.x;

    // ---- Stage x[series, :] into LDS via gfx1250 async global->LDS DMA ----
    // 64 lanes move 16B each (256 floats) while the whole block zeroes the
    // 65x65 count matrix; latency hidden behind s_wait_asynccnt.
    const float* src = x + (size_t)series * LSER;
    if (t < LSER / 4) {
        unsigned ldsoff = (unsigned)(unsigned long long)(lds_f32*)(&xs[t * 4]);
        unsigned voff   = (unsigned)(t * 16);
        asm volatile("global_load_async_to_lds_b128 %0, %1, %2"
                     :: "v"(ldsoff), "v"(voff), "s"(src)
                     : "memory");
    }

    #pragma unroll
    for (int k = 0; k < 17; ++k) {       // 17*256 >= 4225, fixed trip count
        int i = t + k * 256;
        if (i < NC) counts[i] = 0u;
    }
    if (t == 0) { sse[0] = LSER; sse[1] = -1; }

    asm volatile("s_wait_asynccnt 0" ::: "memory");
    __syncthreads();

    // ---- Non-zero run [start, end] (contiguous by construction) ----
    float v = xs[t];
    if (v != 0.0f) {          // float compare: -0.0 counts as zero, like reference
        atomicMin(&sse[0], t);
        atomicMax(&sse[1], t);
    }
    __syncthreads();

    const int start = sse[0];
    const int end   = sse[1];
    const int cntv  = (end >= start) ? (end - start + 1) : 0;
    const int vl    = (cntv > 0) ? cntv : 1;             // clip(valid_len, 1, .)
    const bool validt = (t >= start) && (t <= end);

    // ---- 64-bit sortable key: stable argsort == strict '<' on (bits, idx) ----
    unsigned ub = __float_as_uint(v);
    unsigned sb = (ub & 0x80000000u) ? ~ub : (ub | 0x80000000u);
    const unsigned long long key =
        validt ? (((unsigned long long)sb << 8) | (unsigned)t)
               : 0xFFFFFFFFFFFFFFFFull;   // invalid -> +inf sentinel (> any valid key)
    keys[t] = key;
    __syncthreads();

    // ---- Rank by counting: per pair of elements, one ds_load_b128 broadcast
    //      + two (v_cmp_lt_u64 + carry-add). 128 iterations, fixed trip. ----
    int rank = 0;
    {
        const unsigned long long myk = key;
        const ulonglong2* k2 = (const ulonglong2*)keys;
        #pragma unroll 8
        for (int s = 0; s < LSER / 2; ++s) {
            ulonglong2 kk = k2[s];
            rank += (kk.x < myk) ? 1 : 0;
            rank += (kk.y < myk) ? 1 : 0;
        }
    }

    // ---- Quantize: same f32 op order + truncation as reference ----
    int bin = 0;
    if (validt) {
        float fb = ((float)rank / (float)vl) * 65.0f;
        int b = (int)fb;
        b = (b < 0) ? 0 : b;
        b = (b > NB - 1) ? (NB - 1) : b;
        bin = b;
    }
    binsS[t] = (unsigned char)bin;

    // Normalization LUT: counts are integers in [0, 255]; (float)c / norm here
    // is the exact same IEEE op the reference applies per element.
    const float norm = (float)(((vl - 1) > 1) ? (vl - 1) : 1);
    lut[t] = (float)t / norm;
    __syncthreads();

    // ---- Transition histogram: up to 255 scatter-adds into LDS ----
    if ((t >= start) && (t < end)) {
        unsigned idx = (unsigned)binsS[t] * (unsigned)NB + (unsigned)binsS[t + 1];
        atomicAdd(&counts[idx], 1u);
    }
    __syncthreads();

    // ---- Write-out: LUT lookup instead of 4225 IEEE divides per block;
    //      non-temporal stores (277 MB stream, never re-read, > L2). ----
    float* o = out + (size_t)series * NC;
    #pragma unroll
    for (int k = 0; k < 17; ++k) {
        int i = t + k * 256;
        if (i < NC) {
            __builtin_nontemporal_store(lut[counts[i]], &o[i]);
        }
    }
}

extern "C" void kernel_launch(void* const* d_in, const int* in_sizes, int n_in,
                              void* d_out, int out_size, void* d_ws, size_t ws_size,
                              hipStream_t stream) {
    const float* x = (const float*)d_in[0];
    float* out = (float*)d_out;
    const int nseries = in_sizes[0] / LSER;   // 4096*4 = 16384
    mtf_kernel<<<dim3(nseries), dim3(256), 0, stream>>>(x, out);
    (void)n_in; (void)out_size; (void)d_ws; (void)ws_size;
}